// DiffeqSolver_48137993453878
// MI455X (gfx1250) — compile-verified
//
#include <hip/hip_runtime.h>

typedef float v2f __attribute__((ext_vector_type(2)));
typedef float v8f __attribute__((ext_vector_type(8)));

#define NBATCH   64
#define PTS      325
#define LAT      64
#define HID      128
#define TSTEPS   24
#define TILES    21          // ceil(325/16)
#define SUBSTEPS 4
#define STG_STRIDE 18        // row stride (floats): even -> 8B-aligned v2f frag reads
#define STG_PER_WAVE (16 * STG_STRIDE)

// Hardware tanh (gfx1250 V_TANH_F32 trans op) when the builtin exists;
// co-executes with the WMMA pipe instead of a ~6-op VALU polynomial.
#if __has_builtin(__builtin_amdgcn_tanhf)
__device__ __forceinline__ float fast_tanh(float x) {
    return __builtin_amdgcn_tanhf(x);
}
#else
__device__ __forceinline__ float fast_tanh(float x) { return tanhf(x); }
#endif

// D = A(16x4) * B(4x16) + C, fp32 matrix pipe (CDNA5 WMMA)
__device__ __forceinline__ v8f wmma4(v2f a, v2f b, v8f c) {
    return __builtin_amdgcn_wmma_f32_16x16x4_f32(false, a, false, b,
                                                 (short)0, c, false, false);
}

// dy = tanh(y*W1 + b1)*W2 + b2 for one 16x64 tile held in C-layout (4 x v8f).
// Layout conversion C->A goes through a wave-private LDS staging tile.
__device__ __forceinline__ void f_eval(const v8f y[4], v8f dy[4],
                                       float* __restrict__ stg,
                                       const float* __restrict__ w1p,
                                       const float* __restrict__ w2p,
                                       const float* __restrict__ b1v,
                                       const float* __restrict__ b2v,
                                       int lane) {
    const int row  = lane & 15;
    const int hsel = lane >> 4;    // which 16-lane half
    const int coff = hsel * 2;     // A-frag K offset for upper half

    // ---- extract all 16 A-fragments (16x4) of y via LDS staging ----
    v2f yA[16];
#pragma unroll
    for (int jy = 0; jy < 4; ++jy) {
#pragma unroll
        for (int r = 0; r < 8; ++r)
            stg[(r + hsel * 8) * STG_STRIDE + row] = y[jy][r];
#pragma unroll
        for (int lc = 0; lc < 4; ++lc)
            yA[jy * 4 + lc] =
                *(const v2f*)(stg + row * STG_STRIDE + 4 * lc + coff);
    }

    // dy accumulators start at bias b2 (splat per output column)
    v8f acc[4];
#pragma unroll
    for (int j = 0; j < 4; ++j) {
#pragma unroll
        for (int r = 0; r < 8; ++r) acc[j][r] = b2v[j];
    }

    // One 16-wide tile of H at a time: GEMM1 -> tanh -> stage -> GEMM2.
    // Unroll by 2: two independent GEMM1 accumulator chains for the
    // scheduler to interleave (A-fragments shared between both).
#pragma unroll 2
    for (int g = 0; g < 8; ++g) {
        v8f h;
#pragma unroll
        for (int r = 0; r < 8; ++r) h[r] = b1v[g];

        const float* wb = w1p + g * 64;
#pragma unroll
        for (int kc = 0; kc < 16; ++kc) {
            v2f bf = *(const v2f*)(wb + kc * 8 * 64 + lane * 2);
            h = wmma4(yA[kc], bf, h);
        }
#pragma unroll
        for (int r = 0; r < 8; ++r) h[r] = fast_tanh(h[r]);

        // stage H tile, re-read as A-fragments for GEMM2
#pragma unroll
        for (int r = 0; r < 8; ++r)
            stg[(r + hsel * 8) * STG_STRIDE + row] = h[r];
#pragma unroll
        for (int lc = 0; lc < 4; ++lc) {
            v2f hA = *(const v2f*)(stg + row * STG_STRIDE + 4 * lc + coff);
            const int kc2 = 4 * g + lc;
#pragma unroll
            for (int j = 0; j < 4; ++j) {
                v2f bf2 = *(const v2f*)(w2p + (kc2 * 4 + j) * 64 + lane * 2);
                acc[j] = wmma4(hA, bf2, acc[j]);
            }
        }
    }
#pragma unroll
    for (int j = 0; j < 4; ++j) dy[j] = acc[j];
}

// Rewrite W1[64][128], W2[128][64] into WMMA B-fragment order:
//   W1: 128 blocks (kc 0..15, g 0..7), W2: 128 blocks (kc 0..31, j 0..3),
//   each block = 64 floats: lane L holds (K=2*half+which rows, col 16*t + L%16).
__global__ void prepack_kernel(const float* __restrict__ W1,
                               const float* __restrict__ W2,
                               float* __restrict__ wpack) {
    int idx    = blockIdx.x * blockDim.x + threadIdx.x;  // 0..16383
    int region = idx >> 13;
    int rem    = idx & 8191;
    int fb     = rem >> 6;
    int pos    = rem & 63;
    int lp     = pos >> 1;
    int which  = pos & 1;
    int r      = lp & 15;
    int h2     = lp >> 4;
    float v;
    if (region == 0) {
        int kc = fb >> 3, g = fb & 7;
        int k   = 4 * kc + 2 * h2 + which;
        int col = 16 * g + r;
        v = W1[k * HID + col];
    } else {
        int kc = fb >> 2, j = fb & 3;
        int k   = 4 * kc + 2 * h2 + which;
        int col = 16 * j + r;
        v = W2[k * LAT + col];
    }
    wpack[idx] = v;
}

__global__ __launch_bounds__(256) void ode_rk4_kernel(
    const float* __restrict__ y0g, const float* __restrict__ ts,
    const float* __restrict__ b1, const float* __restrict__ b2,
    const float* __restrict__ wpack, float* __restrict__ out) {
    __shared__ float stage_all[8 * STG_PER_WAVE];
    const int lane  = threadIdx.x & 31;
    const int wslot = threadIdx.x >> 5;
    float* stg = stage_all + wslot * STG_PER_WAVE;

    const int wid   = blockIdx.x * 8 + wslot;   // 1344 waves total
    const int batch = wid / TILES;
    const int tile  = wid % TILES;

    const int row  = lane & 15;
    const int hsel = lane >> 4;

    const float* w1p = wpack;
    const float* w2p = wpack + 128 * 64;

    float b1v[8], b2v[4];
#pragma unroll
    for (int g = 0; g < 8; ++g) b1v[g] = b1[16 * g + row];
#pragma unroll
    for (int j = 0; j < 4; ++j) b2v[j] = b2[16 * j + row];

    // load y0 tile in C layout (rows = points, cols = latent dims)
    v8f y[4];
#pragma unroll
    for (int j = 0; j < 4; ++j) {
#pragma unroll
        for (int r = 0; r < 8; ++r) {
            int m = r + 8 * hsel;
            int p = tile * 16 + m;
            int d = 16 * j + row;
            y[j][r] = (p < PTS) ? y0g[(size_t)(batch * PTS + p) * LAT + d]
                                : 0.0f;
        }
    }

    auto write_out = [&](int t) {
#pragma unroll
        for (int j = 0; j < 4; ++j) {
#pragma unroll
            for (int r = 0; r < 8; ++r) {
                int m = r + 8 * hsel;
                int p = tile * 16 + m;
                if (p < PTS) {
                    int d = 16 * j + row;
                    out[((size_t)(batch * PTS + p) * LAT + d) * TSTEPS + t] =
                        y[j][r];
                }
            }
        }
    };

    // odeint semantics: first output time carries y0
    write_out(0);

    float tprev = ts[batch];  // ts[0][batch]
#pragma unroll 1
    for (int st = 1; st < TSTEPS; ++st) {
        float tcur = ts[st * NBATCH + batch];
        float hh   = (tcur - tprev) * (1.0f / (float)SUBSTEPS);
        tprev = tcur;
#pragma unroll 1
        for (int s = 0; s < SUBSTEPS; ++s) {
            v8f k[4], ks[4], yt[4];
            f_eval(y, k, stg, w1p, w2p, b1v, b2v, lane);   // k1
#pragma unroll
            for (int j = 0; j < 4; ++j) {
                ks[j] = k[j];
                yt[j] = y[j] + (hh * 0.5f) * k[j];
            }
            f_eval(yt, k, stg, w1p, w2p, b1v, b2v, lane);  // k2
#pragma unroll
            for (int j = 0; j < 4; ++j) {
                ks[j] += 2.0f * k[j];
                yt[j] = y[j] + (hh * 0.5f) * k[j];
            }
            f_eval(yt, k, stg, w1p, w2p, b1v, b2v, lane);  // k3
#pragma unroll
            for (int j = 0; j < 4; ++j) {
                ks[j] += 2.0f * k[j];
                yt[j] = y[j] + hh * k[j];
            }
            f_eval(yt, k, stg, w1p, w2p, b1v, b2v, lane);  // k4
#pragma unroll
            for (int j = 0; j < 4; ++j)
                y[j] += (hh * (1.0f / 6.0f)) * (ks[j] + k[j]);
        }
        write_out(st);
    }
}

extern "C" void kernel_launch(void* const* d_in, const int* in_sizes, int n_in,
                              void* d_out, int out_size, void* d_ws,
                              size_t ws_size, hipStream_t stream) {
    const float* first_point = (const float*)d_in[0];
    const float* ts          = (const float*)d_in[1];
    const float* W1          = (const float*)d_in[2];
    const float* b1          = (const float*)d_in[3];
    const float* W2          = (const float*)d_in[4];
    const float* b2          = (const float*)d_in[5];
    float* out   = (float*)d_out;
    float* wpack = (float*)d_ws;  // 16384 floats = 64 KB

    prepack_kernel<<<64, 256, 0, stream>>>(W1, W2, wpack);

    const int nblocks = (NBATCH * TILES) / 8;  // 1344 waves / 8 per block = 168
    ode_rk4_kernel<<<nblocks, 256, 0, stream>>>(first_point, ts, b1, b2, wpack,
                                                out);
}